// StructureDecoder_73933567033959
// MI455X (gfx1250) — compile-verified
//
#include <hip/hip_runtime.h>
#include <hip/hip_bf16.h>
#include <math.h>

// ---------------------------------------------------------------------------
// StructureDecoder for MI455X (gfx1250, wave32).
// Only layer L-1 = 2 affects the output (X is overwritten each layer and the
// per-layer inputs R, t, node_emb, edge_emb are loop-invariant), so we compute
// exactly one layer. The big GEMM (edge_emb[E,128] x Wcat[128,16]) runs on
// V_WMMA_F32_16X16X4_F32 (full f32 precision), two tiles per wave with dual
// accumulators, and a pre-swizzled B matrix so each B fragment is one b64.
// ---------------------------------------------------------------------------

typedef float v2f __attribute__((ext_vector_type(2)));
typedef float v4f __attribute__((ext_vector_type(4)));
typedef float v8f __attribute__((ext_vector_type(8)));

#define NN 20000
#define EE 640000
#define HH 128
#define LAYER 2
#define D_SCALE 10.0f

// ---- order-preserving float <-> uint encoding for atomicMax-based segmax ---
__device__ __forceinline__ unsigned fenc(float f) {
  unsigned u = __float_as_uint(f);
  return (u >> 31) ? ~u : (u | 0x80000000u);
}
__device__ __forceinline__ float fdec(unsigned x) {
  return __uint_as_float((x >> 31) ? (x ^ 0x80000000u) : ~x);
}

// ---- 3x3 helpers ------------------------------------------------------------
__device__ __forceinline__ float det3(const float* m) {
  return m[0]*(m[4]*m[8]-m[5]*m[7])
       - m[1]*(m[3]*m[8]-m[5]*m[6])
       + m[2]*(m[3]*m[7]-m[4]*m[6]);
}
// cofactor matrix: X^{-T} = C / det(X)
__device__ __forceinline__ void cof3(const float* m, float* c) {
  c[0] =  (m[4]*m[8]-m[5]*m[7]);
  c[1] = -(m[3]*m[8]-m[5]*m[6]);
  c[2] =  (m[3]*m[7]-m[4]*m[6]);
  c[3] = -(m[1]*m[8]-m[2]*m[7]);
  c[4] =  (m[0]*m[8]-m[2]*m[6]);
  c[5] = -(m[0]*m[7]-m[1]*m[6]);
  c[6] =  (m[1]*m[5]-m[2]*m[4]);
  c[7] = -(m[0]*m[5]-m[2]*m[3]);
  c[8] =  (m[0]*m[4]-m[1]*m[3]);
}

// Nearest proper rotation via scaled Higham polar iteration.
__device__ void polar3(const float* M, float* R) {
  float X[9];
  float d0  = det3(M);
  float sgn = (d0 < 0.f) ? -1.f : 1.f;
#pragma unroll
  for (int j = 0; j < 9; ++j) X[j] = sgn * M[j];
#pragma unroll
  for (int it = 0; it < 8; ++it) {
    float d = det3(X);
    if (fabsf(d) < 1e-12f) { X[0] += 1e-4f; X[4] += 1e-4f; X[8] += 1e-4f; d = det3(X); }
    float C[9];
    cof3(X, C);
    float rd = 1.f / d;
    float nx = 0.f, nc = 0.f;
#pragma unroll
    for (int j = 0; j < 9; ++j) { C[j] *= rd; nx += X[j]*X[j]; nc += C[j]*C[j]; }
    float g   = sqrtf(sqrtf(nc / nx));     // Frobenius-norm scaling
    float hg  = 0.5f * g;
    float hig = 0.5f / g;
#pragma unroll
    for (int j = 0; j < 9; ++j) X[j] = hg * X[j] + hig * C[j];
  }
#pragma unroll
  for (int j = 0; j < 9; ++j) R[j] = X[j];
}

// ---------------------------------------------------------------------------
// Kernel 0: pack layer-2 weights [Wl|Wq|Wt|Ws] directly in WMMA B-fragment
// order: Wp[step][lane][j] = W[4*step + 2*(lane>>4) + j][lane&15], so the
// GEMM's per-step B fragment is one coalesced b64 load per lane.
// Column map of W[h][c]: 0-1 logit, 2-10 q, 11-13 t, 14-15 s.
// ---------------------------------------------------------------------------
__global__ void prep_kernel(const float* __restrict__ Wl, const float* __restrict__ bl,
                            const float* __restrict__ Wq, const float* __restrict__ bq,
                            const float* __restrict__ Wt, const float* __restrict__ bt,
                            const float* __restrict__ Ws, const float* __restrict__ bs,
                            float* __restrict__ Wp, float* __restrict__ bc) {
  int tid = blockIdx.x * blockDim.x + threadIdx.x;   // 0 .. 2047
  if (tid < 32 * 64) {
    int s    = tid >> 6;           // k-step (0..31)
    int lane = (tid >> 1) & 31;    // consuming lane
    int j    = tid & 1;            // b.x / b.y
    int h    = 4*s + 2*(lane >> 4) + j;   // K index (0..127)
    int c    = lane & 15;                 // output column
    float v;
    if (c < 2)       v = Wl[(LAYER*HH + h)*2 + c];
    else if (c < 11) v = Wq[(LAYER*HH + h)*9 + (c-2)];
    else if (c < 14) v = Wt[(LAYER*HH + h)*3 + (c-11)];
    else             v = Ws[(LAYER*HH + h)*2 + (c-14)];
    Wp[tid] = v;
  }
  if (tid < 16) {
    int c = tid;
    float v;
    if (c < 2)       v = bl[LAYER*2 + c];
    else if (c < 11) v = bq[LAYER*9 + (c-2)];
    else if (c < 14) v = bt[LAYER*3 + (c-11)];
    else             v = bs[LAYER*2 + (c-14)];
    bc[c] = v;
  }
}

// ---------------------------------------------------------------------------
// Kernel 1: OUT16[e][0:16] = edge_emb[e] @ Wcat + bcat  via WMMA f32 16x16x4.
// Two 16-edge tiles per wave (dual accumulators -> two independent WMMA
// chains); K loop 128 in steps of 4.
// A (16x4 f32): lanes 0-15 hold M=lane, K={k,k+1}; lanes 16-31 hold K={k+2,k+3}
// B (4x16):     pre-swizzled Wp, one b64 per lane per step
// C (16x16):    VGPR j: lanes0-15 -> row j, lanes16-31 -> row j+8, col = lane%16
// ---------------------------------------------------------------------------
__global__ void edge_gemm_kernel(const float* __restrict__ emb,
                                 const float* __restrict__ Wp,
                                 const float* __restrict__ bc,
                                 float* __restrict__ out16) {
  int wave = (blockIdx.x * blockDim.x + threadIdx.x) >> 5;
  int lane = threadIdx.x & 31;
  int row0 = wave * 32;                  // two 16-row tiles
  if (row0 >= EE) return;                // wave-uniform

  int m     = lane & 15;                 // M row (A) / N col (B,C)
  int khalf = (lane >> 4) * 2;           // 0 or 2
  const float* a0 = emb + (size_t)(row0 + m) * HH + khalf;
  const float* a1 = a0 + (size_t)16 * HH;
  const float* bp = Wp + lane * 2;

  v8f c0 = {}, c1 = {};
#pragma unroll
  for (int s = 0; s < 32; ++s) {
    v2f b = *(const v2f*)(bp + s * 64);
    v2f x0, x1;
    x0.x = a0[s*4 + 0];  x0.y = a0[s*4 + 1];
    x1.x = a1[s*4 + 0];  x1.y = a1[s*4 + 1];
    c0 = __builtin_amdgcn_wmma_f32_16x16x4_f32(false, x0, false, b,
                                               (short)0, c0, false, false);
    c1 = __builtin_amdgcn_wmma_f32_16x16x4_f32(false, x1, false, b,
                                               (short)0, c1, false, false);
  }

  float bias   = bc[m];
  int   rbase0 = row0 + ((lane >> 4) << 3);
  int   rbase1 = rbase0 + 16;
#pragma unroll
  for (int j = 0; j < 8; ++j) {
    out16[(size_t)(rbase0 + j) * 16 + m] = c0[j] + bias;
    out16[(size_t)(rbase1 + j) * 16 + m] = c1[j] + bias;
  }
}

// ---------------------------------------------------------------------------
// Kernel 2: segment-max of the two logits over src nodes (encoded atomicMax).
// ---------------------------------------------------------------------------
__global__ void pass1_kernel(const float* __restrict__ out16,
                             const int* __restrict__ eidx,
                             unsigned* __restrict__ mx) {
  int e = blockIdx.x * blockDim.x + threadIdx.x;
  if (e >= EE) return;
  int s = eidx[e];
  v2f l = *(const v2f*)(out16 + (size_t)e * 16);
  atomicMax(&mx[2*s + 0], fenc(l.x));
  atomicMax(&mx[2*s + 1], fenc(l.y));
}

// ---------------------------------------------------------------------------
// Kernel 3: per-edge geometry + exp + atomic accumulation into nodes.
// ---------------------------------------------------------------------------
__global__ void pass2_kernel(const float* __restrict__ out16,
                             const int* __restrict__ eidx,
                             const float* __restrict__ Rg,
                             const float* __restrict__ tg,
                             const unsigned* __restrict__ mx,
                             float* __restrict__ den,
                             float* __restrict__ tsum,
                             float* __restrict__ rsum) {
  int e = blockIdx.x * blockDim.x + threadIdx.x;
  if (e >= EE) return;
  int s = eidx[e];          // src (row)
  int d = eidx[EE + e];     // dst (col)

  // 16 per-edge GEMM outputs as four b128 loads (OUT16 region is 16B aligned)
  float o[16];
  const v4f* op = (const v4f*)(out16 + (size_t)e * 16);
#pragma unroll
  for (int q4 = 0; q4 < 4; ++q4) {
    v4f v = op[q4];
    o[q4*4 + 0] = v.x; o[q4*4 + 1] = v.y; o[q4*4 + 2] = v.z; o[q4*4 + 3] = v.w;
  }

  float l0 = o[0], l1 = o[1];
  float s0 = 1.f / (1.f + expf(-o[14]));
  float s1 = 1.f / (1.f + expf(-o[15]));

  float Rr[9], Rc[9];
#pragma unroll
  for (int j = 0; j < 9; ++j) { Rr[j] = Rg[(size_t)s*9 + j]; Rc[j] = Rg[(size_t)d*9 + j]; }
  float dt[3];
#pragma unroll
  for (int i = 0; i < 3; ++i) dt[i] = tg[(size_t)s*3 + i] - tg[(size_t)d*3 + i];

  // q = s0 * (Rc^T @ Rr) + (1-s0) * q_lin
  float q[9];
#pragma unroll
  for (int i = 0; i < 3; ++i)
#pragma unroll
    for (int k = 0; k < 3; ++k) {
      float rji = Rc[0*3+i]*Rr[0*3+k] + Rc[1*3+i]*Rr[1*3+k] + Rc[2*3+i]*Rr[2*3+k];
      q[i*3+k] = s0 * rji + (1.f - s0) * o[2 + i*3 + k];
    }
  // t_ts = s1 * Rc^T (t_s - t_d) + (1-s1) * D_SCALE * t_lin
  float tts[3];
#pragma unroll
  for (int i = 0; i < 3; ++i) {
    float tji = Rc[0*3+i]*dt[0] + Rc[1*3+i]*dt[1] + Rc[2*3+i]*dt[2];
    tts[i] = s1 * tji + (1.f - s1) * D_SCALE * o[11 + i];
  }

  float Rts[9];
  polar3(q, Rts);

  // R_pred = Rc @ R_ts ; t_pred = t_d + Rc @ t_ts
  float Rp[9], tp[3];
#pragma unroll
  for (int i = 0; i < 3; ++i) {
#pragma unroll
    for (int k = 0; k < 3; ++k)
      Rp[i*3+k] = Rc[i*3+0]*Rts[0*3+k] + Rc[i*3+1]*Rts[1*3+k] + Rc[i*3+2]*Rts[2*3+k];
    tp[i] = tg[(size_t)d*3 + i] + Rc[i*3+0]*tts[0] + Rc[i*3+1]*tts[1] + Rc[i*3+2]*tts[2];
  }

  float ex0 = expf(l0 - fdec(mx[2*s + 0]));
  float ex1 = expf(l1 - fdec(mx[2*s + 1]));

  atomicAdd(&den[2*s + 0], ex0);
  atomicAdd(&den[2*s + 1], ex1);
#pragma unroll
  for (int i = 0; i < 3; ++i) atomicAdd(&tsum[3*s + i], ex0 * tp[i]);
#pragma unroll
  for (int j = 0; j < 9; ++j) atomicAdd(&rsum[9*s + j], ex1 * Rp[j]);
}

// ---------------------------------------------------------------------------
// Kernel 4: per-node rotation averaging + atoms GEMV + write X.
// R_avg is scale-invariant, so rsum never needs dividing by den.
// ---------------------------------------------------------------------------
__global__ void node_kernel(const float* __restrict__ ne,
                            const float* __restrict__ Wa,
                            const float* __restrict__ ba,
                            const float* __restrict__ den,
                            const float* __restrict__ tsum,
                            const float* __restrict__ rsum,
                            float* __restrict__ Xout) {
  __shared__ float sWa[HH * 12];
  __shared__ float sba[12];
  const float* Wa2 = Wa + (size_t)LAYER * HH * 12;
  for (int i = threadIdx.x; i < HH * 12; i += blockDim.x) sWa[i] = Wa2[i];
  if (threadIdx.x < 12) sba[threadIdx.x] = ba[LAYER*12 + threadIdx.x];
  __syncthreads();

  int n = blockIdx.x * blockDim.x + threadIdx.x;
  if (n >= NN) return;

  float Rm[9];
  float nrm = 0.f;
#pragma unroll
  for (int j = 0; j < 9; ++j) { Rm[j] = rsum[(size_t)n*9 + j]; nrm += fabsf(Rm[j]); }
  float Ravg[9];
  if (nrm < 1e-20f) {                       // isolated node: identity rotation
#pragma unroll
    for (int j = 0; j < 9; ++j) Ravg[j] = 0.f;
    Ravg[0] = Ravg[4] = Ravg[8] = 1.f;
  } else {
    polar3(Rm, Ravg);
  }

  float d0 = den[2*n + 0];
  float ta[3];
#pragma unroll
  for (int i = 0; i < 3; ++i) ta[i] = (d0 > 0.f) ? tsum[3*n + i] / d0 : 0.f;

  float atoms[12];
#pragma unroll
  for (int c = 0; c < 12; ++c) atoms[c] = sba[c];
  const float* row = ne + (size_t)n * HH;
  for (int h = 0; h < HH; ++h) {
    float v = row[h];
#pragma unroll
    for (int c = 0; c < 12; ++c) atoms[c] += v * sWa[h*12 + c];
  }

#pragma unroll
  for (int a = 0; a < 4; ++a)
#pragma unroll
    for (int i = 0; i < 3; ++i)
      Xout[((size_t)n*4 + a)*3 + i] =
          Ravg[i*3+0]*atoms[a*3+0] + Ravg[i*3+1]*atoms[a*3+1] +
          Ravg[i*3+2]*atoms[a*3+2] + ta[i];
}

// ---------------------------------------------------------------------------
extern "C" void kernel_launch(void* const* d_in, const int* in_sizes, int n_in,
                              void* d_out, int out_size, void* d_ws, size_t ws_size,
                              hipStream_t stream) {
  (void)in_sizes; (void)n_in; (void)out_size; (void)ws_size;

  const float* Rg  = (const float*)d_in[1];
  const float* tg  = (const float*)d_in[2];
  const float* ne  = (const float*)d_in[3];
  const float* ee  = (const float*)d_in[4];
  const int*   eix = (const int*)d_in[5];
  const float* Wl  = (const float*)d_in[7];
  const float* bl  = (const float*)d_in[8];
  const float* Wq  = (const float*)d_in[9];
  const float* bq  = (const float*)d_in[10];
  const float* Wt  = (const float*)d_in[11];
  const float* bt  = (const float*)d_in[12];
  const float* Ws  = (const float*)d_in[13];
  const float* bs  = (const float*)d_in[14];
  const float* Wa  = (const float*)d_in[15];
  const float* ba  = (const float*)d_in[16];
  float* out = (float*)d_out;
  float* ws  = (float*)d_ws;

  // Workspace layout (float elements)
  const size_t WPK   = 0;                   // 32*64 pre-swizzled B fragments
  const size_t BCAT  = WPK + 32*64;         // 16
  const size_t MAXE  = BCAT + 16;           // 2*N (uint-encoded)
  const size_t DEN   = MAXE + 2*(size_t)NN; // 2*N
  const size_t TSUM  = DEN  + 2*(size_t)NN; // 3*N
  const size_t RSUM  = TSUM + 3*(size_t)NN; // 9*N
  const size_t OUT16 = RSUM + 9*(size_t)NN; // 16*E  (~41 MB, 16B aligned)

  // Zero node accumulators; encoded 0 acts as -inf for the segmax.
  hipMemsetAsync(ws + MAXE, 0, (size_t)16 * NN * sizeof(float), stream);

  prep_kernel<<<8, 256, 0, stream>>>(Wl, bl, Wq, bq, Wt, bt, Ws, bs,
                                     ws + WPK, ws + BCAT);

  // EE/32 = 20000 waves, 8 waves (256 threads) per block
  edge_gemm_kernel<<<(EE/32 + 7) / 8, 256, 0, stream>>>(ee, ws + WPK, ws + BCAT,
                                                        ws + OUT16);

  pass1_kernel<<<(EE + 255) / 256, 256, 0, stream>>>(ws + OUT16, eix,
                                                     (unsigned*)(ws + MAXE));

  pass2_kernel<<<(EE + 255) / 256, 256, 0, stream>>>(ws + OUT16, eix, Rg, tg,
                                                     (const unsigned*)(ws + MAXE),
                                                     ws + DEN, ws + TSUM, ws + RSUM);

  node_kernel<<<(NN + 255) / 256, 256, 0, stream>>>(ne, Wa, ba, ws + DEN,
                                                    ws + TSUM, ws + RSUM, out);
}